// ProxyGML_28037546508957
// MI455X (gfx1250) — compile-verified
//
#include <hip/hip_runtime.h>
#include <math.h>

#define B_    1024
#define DIM_  512
#define C_    1000
#define N_    8
#define CN_   8000
#define TOPK_ 400

typedef __attribute__((ext_vector_type(2))) float v2f;
typedef __attribute__((ext_vector_type(8))) float v8f;

#define LDA  516  // A pitch: 512 + 4 -> 16 lanes at stride 516 hit 16 distinct banks
#define KC   16   // K-chunk staged per double-buffer step
#define BPAD 72   // B pitch: half-waves (2-row offset = 144 dw -> +16 banks) disjoint

// ---------------------------------------------------------------------------
// centers = proxies / max(||col||, 1e-12), column-wise over DIM
// ---------------------------------------------------------------------------
__global__ __launch_bounds__(256) void k_normalize(const float* __restrict__ P,
                                                   float* __restrict__ Cn) {
    int j = blockIdx.x * blockDim.x + threadIdx.x;
    if (j >= CN_) return;
    float s = 0.f;
    for (int d = 0; d < DIM_; ++d) {
        float v = P[(size_t)d * CN_ + j];
        s += v * v;
    }
    float r = 1.0f / fmaxf(sqrtf(s), 1e-12f);
    for (int d = 0; d < DIM_; ++d)
        Cn[(size_t)d * CN_ + j] = P[(size_t)d * CN_ + j] * r;
}

// ---------------------------------------------------------------------------
// centers_cls[d,c] = sum_n centers[d, c*8+n]
// ---------------------------------------------------------------------------
__global__ __launch_bounds__(256) void k_ccls(const float* __restrict__ Cn,
                                              float* __restrict__ Ccls) {
    int idx = blockIdx.x * blockDim.x + threadIdx.x;
    if (idx >= DIM_ * C_) return;
    int d = idx / C_, c = idx % C_;
    const float* p = Cn + (size_t)d * CN_ + (size_t)c * N_;
    float s = 0.f;
#pragma unroll
    for (int n = 0; n < N_; ++n) s += p[n];
    Ccls[idx] = s;
}

// ---------------------------------------------------------------------------
// sim = x (1024x512) @ centers (512x8000)   [V_WMMA_F32_16X16X4_F32]
// block = 4 waves; tile 16(M) x 64(N); A in LDS once; B double-buffered in LDS
// via coalesced b128 fills (16 K-rows x 64 N per chunk)
// ---------------------------------------------------------------------------
__global__ __launch_bounds__(128) void k_gemm_sim(const float* __restrict__ X,
                                                  const float* __restrict__ Cn,
                                                  float* __restrict__ Sim) {
    __shared__ float As[16 * LDA];
    __shared__ float Bs[2][KC][BPAD];
    const int m0   = blockIdx.y * 16;
    const int nblk = blockIdx.x * 64;
    const int t    = threadIdx.x;

    // A: 16 consecutive rows of X are contiguous -> fully coalesced fill
    for (int i = t; i < 16 * DIM_; i += 128) {
        int r = i >> 9, c = i & 511;
        As[r * LDA + c] = X[(size_t)(m0 + r) * DIM_ + c];
    }

    // B chunk fill: 16 rows x 64 cols = 256 float4, 2 per thread (b128 loads)
    const int fr0 = t >> 4,          fc0 = (t & 15) * 4;
    const int fr1 = (t + 128) >> 4,  fc1 = fc0;  // second rep: rows 8..15
    // prime buffer 0 (kbase = 0)
    *(float4*)&Bs[0][fr0][fc0] = *(const float4*)(Cn + (size_t)fr0 * CN_ + nblk + fc0);
    *(float4*)&Bs[0][fr1][fc1] = *(const float4*)(Cn + (size_t)fr1 * CN_ + nblk + fc1);
    __syncthreads();

    const int wave = t >> 5, lane = t & 31;
    const int half = lane >> 4, q = lane & 15;
    const int bq   = wave * 16 + q;  // column within block's 64

    v8f acc = {};
    const float* Arow = As + q * LDA;
    const int NCH = DIM_ / KC;  // 32 chunks
    for (int ch = 0; ch < NCH; ++ch) {
        const int cur = ch & 1;
        if (ch + 1 < NCH) {  // overlap next chunk's global->LDS fill with WMMA
            const int nxt = cur ^ 1;
            const int kb1 = (ch + 1) * KC;
            *(float4*)&Bs[nxt][fr0][fc0] =
                *(const float4*)(Cn + (size_t)(kb1 + fr0) * CN_ + nblk + fc0);
            *(float4*)&Bs[nxt][fr1][fc1] =
                *(const float4*)(Cn + (size_t)(kb1 + fr1) * CN_ + nblk + fc1);
        }
        const int kb = ch * KC;
#pragma unroll
        for (int kk = 0; kk < KC; kk += 4) {
            v2f a, b;
            // A 16x4 frag: lanes 0-15 hold K=0,1 ; lanes 16-31 hold K=2,3
            a[0] = Arow[kb + kk + half * 2];
            a[1] = Arow[kb + kk + half * 2 + 1];
            // B 4x16 frag from LDS (compiler merges to ds_load_2addr)
            b[0] = Bs[cur][kk + half * 2][bq];
            b[1] = Bs[cur][kk + half * 2 + 1][bq];
            acc = __builtin_amdgcn_wmma_f32_16x16x4_f32(false, a, false, b,
                                                        (short)0, acc, false, false);
        }
        __syncthreads();
    }

    // C/D layout: VGPR r -> M=r (lanes 0-15) / M=8+r (lanes 16-31), N=lane%16
    const int col  = nblk + bq;
    const int rowb = m0 + half * 8;
#pragma unroll
    for (int r = 0; r < 8; ++r)
        Sim[(size_t)(rowb + r) * CN_ + col] = acc[r];
}

// ---------------------------------------------------------------------------
// clogits = centers.T (8000x512) @ centers_cls (512x1000)   [WMMA]
// Same structure; N=1000 needs guards on B fill and C store.
// ---------------------------------------------------------------------------
__global__ __launch_bounds__(128) void k_gemm_reg(const float* __restrict__ Cn,
                                                  const float* __restrict__ Ccls,
                                                  float* __restrict__ CL) {
    __shared__ float As[16 * LDA];
    __shared__ float Bs[2][KC][BPAD];
    const int i0   = blockIdx.y * 16;  // row block in CN
    const int nblk = blockIdx.x * 64;
    const int t    = threadIdx.x;

    // As[m][k] = centers[k*CN + i0+m]; consecutive t -> consecutive m (coalesced)
    {
        int m = t & 15, k0 = t >> 4;
        for (int k = k0; k < DIM_; k += 8)
            As[m * LDA + k] = Cn[(size_t)k * CN_ + i0 + m];
    }

    const int fr0 = t >> 4,         fc0 = (t & 15) * 4;
    const int fr1 = (t + 128) >> 4, fc1 = fc0;

    auto fillB = [&](int buf, int kbase) {
        int col0 = nblk + fc0;
        if (col0 + 3 < C_) {
            *(float4*)&Bs[buf][fr0][fc0] =
                *(const float4*)(Ccls + (size_t)(kbase + fr0) * C_ + col0);
            *(float4*)&Bs[buf][fr1][fc1] =
                *(const float4*)(Ccls + (size_t)(kbase + fr1) * C_ + col0);
        } else {
#pragma unroll
            for (int e = 0; e < 4; ++e) {
                int c = col0 + e;
                Bs[buf][fr0][fc0 + e] = (c < C_) ? Ccls[(size_t)(kbase + fr0) * C_ + c] : 0.f;
                Bs[buf][fr1][fc1 + e] = (c < C_) ? Ccls[(size_t)(kbase + fr1) * C_ + c] : 0.f;
            }
        }
    };

    fillB(0, 0);
    __syncthreads();

    const int wave = t >> 5, lane = t & 31;
    const int half = lane >> 4, q = lane & 15;
    const int bq   = wave * 16 + q;

    v8f acc = {};
    const float* Arow = As + q * LDA;
    const int NCH = DIM_ / KC;
    for (int ch = 0; ch < NCH; ++ch) {
        const int cur = ch & 1;
        if (ch + 1 < NCH) fillB(cur ^ 1, (ch + 1) * KC);
        const int kb = ch * KC;
#pragma unroll
        for (int kk = 0; kk < KC; kk += 4) {
            v2f a, b;
            a[0] = Arow[kb + kk + half * 2];
            a[1] = Arow[kb + kk + half * 2 + 1];
            b[0] = Bs[cur][kk + half * 2][bq];
            b[1] = Bs[cur][kk + half * 2 + 1][bq];
            acc = __builtin_amdgcn_wmma_f32_16x16x4_f32(false, a, false, b,
                                                        (short)0, acc, false, false);
        }
        __syncthreads();
    }

    const int col = nblk + bq;
    if (col < C_) {
        const int rowb = i0 + half * 8;
#pragma unroll
        for (int r = 0; r < 8; ++r)
            CL[(size_t)(rowb + r) * C_ + col] = acc[r];
    }
}

// order-preserving uint encoding of float (total order, ascending)
__device__ __forceinline__ unsigned ford(float f) {
    unsigned u = __float_as_uint(f);
    return (u & 0x80000000u) ? ~u : (u | 0x80000000u);
}

// ---------------------------------------------------------------------------
// Per-row: top-400 threshold (binary search on uint codes), masked class
// logits, masked softmax CE; atomic accumulate of per-row loss.
// ---------------------------------------------------------------------------
__global__ __launch_bounds__(256) void k_loss_cls(const float* __restrict__ Sim,
                                                  const int* __restrict__ tgt,
                                                  float* __restrict__ acc) {
    __shared__ float    srow[CN_];  // 32 KB row cache (320 KB/WGP available)
    __shared__ unsigned scnt[256];
    __shared__ float    sred[256];
    __shared__ float    setgt;

    const int b  = blockIdx.x;
    const int t  = threadIdx.x;
    const int tg = tgt[b];

    for (int j = t; j < CN_; j += 256) srow[j] = Sim[(size_t)b * CN_ + j];
    if (t == 0) setgt = 0.f;
    __syncthreads();

    // find max T with count(code >= T) >= TOPK  (code of boosted value)
    unsigned lo = 0u, hi = 0xFFFFFFFFu;
    for (int it = 0; it < 32; ++it) {
        unsigned d   = hi - lo;
        unsigned mid = lo + (d >> 1) + (d & 1u);  // ceil midpoint, overflow-safe
        unsigned cnt = 0;
        for (int j = t; j < CN_; j += 256) {
            float v = srow[j] + (((j >> 3) == tg) ? 1000.0f : 0.0f);
            cnt += (ford(v) >= mid) ? 1u : 0u;
        }
        scnt[t] = cnt;
        __syncthreads();
        for (int s = 128; s > 0; s >>= 1) {
            if (t < s) scnt[t] += scnt[t + s];
            __syncthreads();
        }
        unsigned total = scnt[0];
        __syncthreads();
        if (total >= TOPK_) lo = mid; else hi = mid - 1u;
    }
    const unsigned T = lo;

    // masked class logits -> masked softmax CE
    float denom = 0.f;
    for (int c = t; c < C_; c += 256) {
        float boost = (c == tg) ? 1000.f : 0.f;
        float l = 0.f;
#pragma unroll
        for (int n = 0; n < N_; ++n) {
            float v = srow[c * N_ + n];
            if (ford(v + boost) >= T) l += v;
        }
        float e = (l != 0.f) ? expf(l) : 0.f;  // logits_mask fused
        denom += e;
        if (c == tg) setgt = e;
    }
    sred[t] = denom;
    __syncthreads();
    for (int s = 128; s > 0; s >>= 1) {
        if (t < s) sred[t] += sred[t + s];
        __syncthreads();
    }
    if (t == 0) {
        float p = setgt / (1e-8f + sred[0]);
        atomicAdd(acc, -logf(p + 1e-20f));
    }
}

// ---------------------------------------------------------------------------
// reg: per row i of clogits (8000x1000): logsumexp - row[i/8]; accumulate
// ---------------------------------------------------------------------------
__global__ __launch_bounds__(256) void k_loss_reg(const float* __restrict__ CL,
                                                  float* __restrict__ acc) {
    __shared__ float sred[256];
    const int i = blockIdx.x;
    const int t = threadIdx.x;
    const float* row = CL + (size_t)i * C_;

    float mx = -INFINITY;
    for (int c = t; c < C_; c += 256) mx = fmaxf(mx, row[c]);
    sred[t] = mx;
    __syncthreads();
    for (int s = 128; s > 0; s >>= 1) {
        if (t < s) sred[t] = fmaxf(sred[t], sred[t + s]);
        __syncthreads();
    }
    mx = sred[0];
    __syncthreads();

    float se = 0.f;
    for (int c = t; c < C_; c += 256) se += expf(row[c] - mx);
    sred[t] = se;
    __syncthreads();
    for (int s = 128; s > 0; s >>= 1) {
        if (t < s) sred[t] += sred[t + s];
        __syncthreads();
    }
    if (t == 0) {
        float lse = mx + logf(sred[0]);
        atomicAdd(acc, lse - row[i / N_]);  // -log_softmax at target column
    }
}

__global__ void k_final(const float* __restrict__ accs, float* __restrict__ out) {
    float lc = accs[0] / (float)B_;
    float rg = accs[1] / (float)CN_;
    out[0] = lc + 0.3f * rg;
    out[1] = lc;
}

// ---------------------------------------------------------------------------
extern "C" void kernel_launch(void* const* d_in, const int* in_sizes, int n_in,
                              void* d_out, int out_size, void* d_ws, size_t ws_size,
                              hipStream_t stream) {
    const float* x       = (const float*)d_in[0];
    const float* proxies = (const float*)d_in[1];
    const int*   target  = (const int*)d_in[2];
    float* out = (float*)d_out;

    // workspace layout (floats): centers | centers_cls | sim | clogits | accs
    float* centers = (float*)d_ws;                        // 512*8000  (16 MB)
    float* ccls    = centers + (size_t)DIM_ * CN_;        // 512*1000  ( 2 MB)
    float* sim     = ccls    + (size_t)DIM_ * C_;         // 1024*8000 (32 MB)
    float* clog    = sim     + (size_t)B_   * CN_;        // 8000*1000 (32 MB)
    float* accs    = clog    + (size_t)CN_  * C_;         // 2 accumulators

    hipMemsetAsync(accs, 0, 2 * sizeof(float), stream);

    k_normalize<<<(CN_ + 255) / 256, 256, 0, stream>>>(proxies, centers);
    k_ccls<<<(DIM_ * C_ + 255) / 256, 256, 0, stream>>>(centers, ccls);

    dim3 g1(CN_ / 64, B_ / 16);           // 125 x 64 blocks
    k_gemm_sim<<<g1, 128, 0, stream>>>(x, centers, sim);

    dim3 g2((C_ + 63) / 64, CN_ / 16);    // 16 x 500 blocks
    k_gemm_reg<<<g2, 128, 0, stream>>>(centers, ccls, clog);

    k_loss_cls<<<B_, 256, 0, stream>>>(sim, target, accs + 0);
    k_loss_reg<<<CN_, 256, 0, stream>>>(clog, accs + 1);
    k_final<<<1, 1, 0, stream>>>(accs, out);
}